// Temporal_Heterogeneity_AttentionLayer_43379169689892
// MI455X (gfx1250) — compile-verified
//
#include <hip/hip_runtime.h>
#include <hip/hip_bf16.h>
#include <stddef.h>

// ---------------------------------------------------------------------------
// Problem constants (from reference): B=4, N=170, T=96, D=256, H=8, HD=32,
// TD=24, HIST=3, P=16.  Tokens M = B*N*T = 65280 (multiple of 64).
// ---------------------------------------------------------------------------
#define BB   4
#define NN   170
#define TT   96
#define DD   256
#define HH   8
#define HDD  32
#define TDD  24
#define HIST 3
#define PP   16
#define MTOK (BB * NN * TT)   // 65280

typedef __attribute__((ext_vector_type(2))) float v2f;
typedef __attribute__((ext_vector_type(8))) float v8f;

// ===========================================================================
// Kernel 1: history_pattern = pattern_matrix @ Wp.T + bp   (16,3,24)
// ===========================================================================
__global__ void pattern_proj_kernel(const float* __restrict__ pm,
                                    const float* __restrict__ Wp,
                                    const float* __restrict__ bp,
                                    float* __restrict__ hp) {
    int idx = blockIdx.x * blockDim.x + threadIdx.x;   // p*3*24 + j*24 + d
    if (idx >= PP * HIST * TDD) return;
    int d  = idx % TDD;
    int pj = idx / TDD;
    const float* row = pm + pj * TDD;
    float s = bp[d];
    #pragma unroll
    for (int e = 0; e < TDD; ++e) s += row[e] * Wp[d * TDD + e];
    hp[idx] = s;
}

// ===========================================================================
// Kernel 2: zero a float buffer
// ===========================================================================
__global__ void zero_kernel(float* __restrict__ p, int n) {
    int idx = blockIdx.x * blockDim.x + threadIdx.x;
    if (idx < n) p[idx] = 0.0f;
}

// ===========================================================================
// Kernel 3: sim accumulation over n.
//  sim[b,n,l,p] = (1/3) sum_{j,d} query[b,n,l+j,d<24] * hp[p,j,d]
//  accumulated (atomicAdd) into simacc[b, l+1, p]  (pad=1; t=0,95 stay 0)
// ===========================================================================
__global__ __launch_bounds__(128) void sim_accum_kernel(
        const float* __restrict__ query,
        const float* __restrict__ hp,
        float* __restrict__ simacc) {
    __shared__ float tfs[TT][TDD];             // 96x24 = 9.2 KB
    __shared__ float hps[PP * HIST * TDD];     // 1152  = 4.6 KB
    const int n = blockIdx.x, b = blockIdx.y;
    const float* qg = query + ((size_t)(b * NN + n) * TT) * DD;

    for (int idx = threadIdx.x; idx < TT * TDD; idx += 128) {
        int row = idx / TDD, c = idx % TDD;
        tfs[row][c] = qg[row * DD + c];
    }
    for (int idx = threadIdx.x; idx < PP * HIST * TDD; idx += 128)
        hps[idx] = hp[idx];
    __syncthreads();

    const int L = TT - HIST + 1;               // 94
    for (int idx = threadIdx.x; idx < L * PP; idx += 128) {
        int p = idx % PP;
        int l = idx / PP;
        float s = 0.0f;
        #pragma unroll
        for (int j = 0; j < HIST; ++j) {
            const float* hrow = &hps[(p * HIST + j) * TDD];
            #pragma unroll
            for (int d = 0; d < TDD; ++d) s += tfs[l + j][d] * hrow[d];
        }
        atomicAdd(&simacc[(b * TT + (l + 1)) * PP + p], s * (1.0f / 3.0f));
    }
}

// ===========================================================================
// Kernel 4: w[b,t] = sum_p softmax_p( simacc[b,t,:] / N )   (≈ 1.0)
// ===========================================================================
__global__ void softmax_w_kernel(const float* __restrict__ simacc,
                                 float* __restrict__ wv) {
    int idx = blockIdx.x * blockDim.x + threadIdx.x;   // b*T + t
    if (idx >= BB * TT) return;
    const float* row = simacc + idx * PP;
    float vals[PP];
    float m = -3.402823466e38f;
    #pragma unroll
    for (int p = 0; p < PP; ++p) {
        vals[p] = row[p] * (1.0f / (float)NN);
        m = fmaxf(m, vals[p]);
    }
    float s = 0.0f;
    #pragma unroll
    for (int p = 0; p < PP; ++p) s += __expf(vals[p] - m);
    float wsum = 0.0f;
    #pragma unroll
    for (int p = 0; p < PP; ++p) wsum += __expf(vals[p] - m) / s;
    wv[idx] = wsum;
}

// ===========================================================================
// Kernel 5: Y[M,256] = X[M,256] @ W^T + bias   via V_WMMA_F32_16X16X4_F32
//  Block: 256 threads (8 waves) -> 64x64 output tile, K chunked by 32 in LDS.
//  Wave w computes subtiles (mi = w&3, ni = 2*(w>>2)+{0,1}).
// ===========================================================================
__global__ __launch_bounds__(256) void gemm_bias_wmma(
        const float* __restrict__ X,    // [M, 256]
        const float* __restrict__ W,    // [256, 256] row-major (Y = X W^T + b)
        const float* __restrict__ bias, // [256]
        float* __restrict__ Y) {
    __shared__ float Xs[64][36];        // stride 36 -> 16B-aligned float4 rows
    __shared__ float Ws[64][36];

    const int tid   = threadIdx.x;
    const int lane  = tid & 31;
    const int wave  = tid >> 5;                 // 0..7
    const int mBase = blockIdx.y * 64;
    const int nBase = blockIdx.x * 64;
    const int mi    = wave & 3;                 // M subtile 0..3
    const int ni0   = (wave >> 2) * 2;          // N subtiles ni0, ni0+1

    v8f acc0 = {};
    v8f acc1 = {};

    const int hl = lane >> 4;                   // 0: K={0,1}  1: K={2,3}
    const int r  = lane & 15;

    for (int kb = 0; kb < DD; kb += 32) {
        // cooperative load: 64 rows x 32 cols = 512 float4, 2 per thread
        #pragma unroll
        for (int it = 0; it < 2; ++it) {
            int idx = tid + it * 256;
            int row = idx >> 3, c4 = idx & 7;
            float4 xv = *(const float4*)(X + (size_t)(mBase + row) * DD + kb + c4 * 4);
            float4 wv = *(const float4*)(W + (size_t)(nBase + row) * DD + kb + c4 * 4);
            *(float4*)&Xs[row][c4 * 4] = xv;
            *(float4*)&Ws[row][c4 * 4] = wv;
        }
        __syncthreads();

        #pragma unroll
        for (int k = 0; k < 32; k += 4) {
            const int kk = k + (hl << 1);
            v2f a  = *(const v2f*)&Xs[mi * 16 + r][kk];
            v2f b0 = *(const v2f*)&Ws[(ni0 + 0) * 16 + r][kk];
            v2f b1 = *(const v2f*)&Ws[(ni0 + 1) * 16 + r][kk];
            acc0 = __builtin_amdgcn_wmma_f32_16x16x4_f32(false, a, false, b0,
                                                         (short)0, acc0, false, false);
            acc1 = __builtin_amdgcn_wmma_f32_16x16x4_f32(false, a, false, b1,
                                                         (short)0, acc1, false, false);
        }
        __syncthreads();
    }

    // D layout: VGPR i -> M = i (lanes 0-15) / i+8 (lanes 16-31); N = lane&15
    const int c = lane & 15;
    #pragma unroll
    for (int i = 0; i < 8; ++i) {
        int row = mBase + mi * 16 + i + 8 * hl;
        int col0 = nBase + (ni0 + 0) * 16 + c;
        int col1 = nBase + (ni0 + 1) * 16 + c;
        Y[(size_t)row * DD + col0] = acc0[i] + bias[col0];
        Y[(size_t)row * DD + col1] = acc1[i] + bias[col1];
    }
}

// ===========================================================================
// Kernel 6: attention for one (b, n, h): scores = QK^T/sqrt(32) * mask,
//  softmax, out = attn @ V.  64 threads = 2 waves; each wave owns 3 q-tiles.
// ===========================================================================
__global__ __launch_bounds__(64) void attn_wmma(
        const float* __restrict__ Q, const float* __restrict__ K,
        const float* __restrict__ V, const float* __restrict__ wvec,
        float* __restrict__ CTX) {
    __shared__ float Qs[TT][36];        // 96x32 head tile (stride 36)
    __shared__ float Ks[TT][36];
    __shared__ float Vs[TT][36];
    __shared__ float Sbuf[2][16][100];  // per-wave 16x96 scores
    __shared__ float wvs[TT];

    const int h = blockIdx.x, n = blockIdx.y, b = blockIdx.z;
    const int tid = threadIdx.x, lane = tid & 31, wave = tid >> 5;
    const size_t base = ((size_t)(b * NN + n) * TT) * DD + h * HDD;

    // load 96x32 tiles (8 float4 per row)
    for (int idx = tid; idx < TT * 8; idx += 64) {
        int row = idx >> 3, c4 = idx & 7;
        *(float4*)&Qs[row][c4 * 4] = *(const float4*)(Q + base + (size_t)row * DD + c4 * 4);
        *(float4*)&Ks[row][c4 * 4] = *(const float4*)(K + base + (size_t)row * DD + c4 * 4);
        *(float4*)&Vs[row][c4 * 4] = *(const float4*)(V + base + (size_t)row * DD + c4 * 4);
    }
    for (int idx = tid; idx < TT; idx += 64) wvs[idx] = wvec[b * TT + idx];
    __syncthreads();

    const float scale = 0.1767766952966369f;   // 1/sqrt(32)
    const int hl = lane >> 4;
    const int r  = lane & 15;
    const int c  = lane & 15;

    for (int qt = 0; qt < 3; ++qt) {           // both waves: 3 iterations each
        const int qi = wave + 2 * qt;          // q tile 0..5

        // ---- scores = Q(16x96-slice) @ K^T : 6 N-tiles, 24 K-steps --------
        v8f s[6];
        #pragma unroll
        for (int t = 0; t < 6; ++t) s[t] = (v8f){};
        for (int k = 0; k < TT; k += 4) {
            const int kk = k + (hl << 1);
            v2f a = *(const v2f*)&Qs[qi * 16 + r][kk];
            #pragma unroll
            for (int ni = 0; ni < 6; ++ni) {
                v2f bb = *(const v2f*)&Ks[ni * 16 + r][kk];
                s[ni] = __builtin_amdgcn_wmma_f32_16x16x4_f32(false, a, false, bb,
                                                              (short)0, s[ni], false, false);
            }
        }

        // ---- scale * mask(w_q*w_k), spill to LDS ---------------------------
        #pragma unroll
        for (int ni = 0; ni < 6; ++ni) {
            int col = ni * 16 + c;
            float wk = wvs[col];
            #pragma unroll
            for (int i = 0; i < 8; ++i) {
                int lrow = i + 8 * hl;                       // 0..15 in tile
                float wq = wvs[qi * 16 + lrow];
                Sbuf[wave][lrow][col] = s[ni][i] * scale * wq * wk;
            }
        }
        __syncthreads();

        // ---- row softmax (lanes 0..15 each own one row) --------------------
        if (lane < 16) {
            float m = -3.402823466e38f;
            for (int j = 0; j < TT; ++j) m = fmaxf(m, Sbuf[wave][lane][j]);
            float ssum = 0.0f;
            for (int j = 0; j < TT; ++j) {
                float e = __expf(Sbuf[wave][lane][j] - m);
                Sbuf[wave][lane][j] = e;
                ssum += e;
            }
            float inv = 1.0f / ssum;
            for (int j = 0; j < TT; ++j) Sbuf[wave][lane][j] *= inv;
        }
        __syncthreads();

        // ---- out = attn(16x96) @ V(96x32) : 2 N-tiles, 24 K-steps ----------
        v8f o0 = {};
        v8f o1 = {};
        for (int k = 0; k < TT; k += 4) {
            const int kk = k + (hl << 1);
            v2f a = *(const v2f*)&Sbuf[wave][r][kk];
            v2f b0, b1;
            b0.x = Vs[kk][c];          b0.y = Vs[kk + 1][c];
            b1.x = Vs[kk][16 + c];     b1.y = Vs[kk + 1][16 + c];
            o0 = __builtin_amdgcn_wmma_f32_16x16x4_f32(false, a, false, b0,
                                                       (short)0, o0, false, false);
            o1 = __builtin_amdgcn_wmma_f32_16x16x4_f32(false, a, false, b1,
                                                       (short)0, o1, false, false);
        }

        #pragma unroll
        for (int i = 0; i < 8; ++i) {
            int row = qi * 16 + i + 8 * hl;
            float* dst = CTX + ((size_t)(b * NN + n) * TT + row) * DD + h * HDD;
            dst[c]      = o0[i];
            dst[16 + c] = o1[i];
        }
        __syncthreads();   // Sbuf reused next iteration
    }
}

// ===========================================================================
// Launcher
// ===========================================================================
extern "C" void kernel_launch(void* const* d_in, const int* in_sizes, int n_in,
                              void* d_out, int out_size, void* d_ws, size_t ws_size,
                              hipStream_t stream) {
    const float* query = (const float*)d_in[0];
    const float* key   = (const float*)d_in[1];
    const float* value = (const float*)d_in[2];
    const float* pm    = (const float*)d_in[3];
    const float* Wp    = (const float*)d_in[4];
    const float* bp    = (const float*)d_in[5];
    const float* Wq    = (const float*)d_in[6];
    const float* bq    = (const float*)d_in[7];
    const float* Wk    = (const float*)d_in[8];
    const float* bk    = (const float*)d_in[9];
    const float* Wv    = (const float*)d_in[10];
    const float* bv    = (const float*)d_in[11];
    const float* Wo    = (const float*)d_in[12];
    const float* bo    = (const float*)d_in[13];
    float* out = (float*)d_out;

    // workspace layout (floats); each big buffer is 16B-aligned
    const size_t big = (size_t)MTOK * DD;           // 16,711,680
    float* Qb     = (float*)d_ws;
    float* Kb     = Qb + big;
    float* Vb     = Kb + big;
    float* Cb     = Vb + big;
    float* hp     = Cb + big;                       // 1152
    float* simacc = hp + PP * HIST * TDD;           // 6144
    float* wv     = simacc + BB * TT * PP;          // 384
    (void)ws_size; (void)in_sizes; (void)n_in; (void)out_size;

    // --- mask branch (tiny) -------------------------------------------------
    pattern_proj_kernel<<<(PP * HIST * TDD + 255) / 256, 256, 0, stream>>>(pm, Wp, bp, hp);
    zero_kernel<<<(BB * TT * PP + 255) / 256, 256, 0, stream>>>(simacc, BB * TT * PP);
    sim_accum_kernel<<<dim3(NN, BB), 128, 0, stream>>>(query, hp, simacc);
    softmax_w_kernel<<<(BB * TT + 255) / 256, 256, 0, stream>>>(simacc, wv);

    // --- Q/K/V projections (WMMA f32 GEMM) ----------------------------------
    dim3 ggrid(DD / 64, MTOK / 64);                 // (4, 1020)
    gemm_bias_wmma<<<ggrid, 256, 0, stream>>>(query, Wq, bq, Qb);
    gemm_bias_wmma<<<ggrid, 256, 0, stream>>>(key,   Wk, bk, Kb);
    gemm_bias_wmma<<<ggrid, 256, 0, stream>>>(value, Wv, bv, Vb);

    // --- attention (WMMA) ---------------------------------------------------
    attn_wmma<<<dim3(HH, NN, BB), 64, 0, stream>>>(Qb, Kb, Vb, wv, Cb);

    // --- output projection --------------------------------------------------
    gemm_bias_wmma<<<ggrid, 256, 0, stream>>>(Cb, Wo, bo, out);
}